// Encoder_57655640981587
// MI455X (gfx1250) — compile-verified
//
#include <hip/hip_runtime.h>
#include <hip/hip_bf16.h>

typedef __attribute__((ext_vector_type(2))) float v2f;
typedef __attribute__((ext_vector_type(8))) float v8f;

// ============================================================================
// Farthest point sampling: one 1024-thread workgroup per batch. dist vector in
// workspace (L2-resident). First-index tie-break to match jnp.argmax.
// Coord layout parameterized by strides: handles xyz (B,6,N) channel-major and
// l1_xyz (B,S,3) point-major with one kernel.
// ============================================================================
__global__ void fps_kernel(const float* __restrict__ base, long cb, long cp, long cc,
                           int n, int npoint, int* __restrict__ centIdx,
                           float* __restrict__ newXyz, float* __restrict__ distWS)
{
    __shared__ float rv[1024];
    __shared__ int   ri[1024];
    __shared__ int   sFar;
    const int b   = blockIdx.x;
    const int tid = threadIdx.x;
    const float* P = base + (long)b * cb;
    float* db = distWS + (long)b * n;

    for (int j = tid; j < n; j += 1024) db[j] = 1e10f;
    if (tid == 0) sFar = 0;
    __syncthreads();

    for (int i = 0; i < npoint; ++i) {
        const int far = sFar;
        const float cx = P[cp*far + cc*0];
        const float cy = P[cp*far + cc*1];
        const float cz = P[cp*far + cc*2];
        if (tid == 0) {
            centIdx[b*npoint + i] = far;
            newXyz[((long)b*npoint + i)*3 + 0] = cx;
            newXyz[((long)b*npoint + i)*3 + 1] = cy;
            newXyz[((long)b*npoint + i)*3 + 2] = cz;
        }
        float bestv = -1.0f; int besti = 0;
        for (int j = tid; j < n; j += 1024) {
            const float dx = P[cp*j + cc*0] - cx;
            const float dy = P[cp*j + cc*1] - cy;
            const float dz = P[cp*j + cc*2] - cz;
            const float d  = dx*dx + dy*dy + dz*dz;
            const float nd = fminf(db[j], d);
            db[j] = nd;
            if (nd > bestv) { bestv = nd; besti = j; }   // strict > keeps first index
        }
        rv[tid] = bestv; ri[tid] = besti;
        __syncthreads();
        for (int w = 512; w > 0; w >>= 1) {
            if (tid < w) {
                const float ov = rv[tid + w]; const int oi = ri[tid + w];
                if (ov > rv[tid] || (ov == rv[tid] && oi < ri[tid])) { rv[tid] = ov; ri[tid] = oi; }
            }
            __syncthreads();
        }
        if (tid == 0) sFar = ri[0];
        __syncthreads();
    }
}

// ============================================================================
// Fused ball-query + grouping for all 3 radii: one wave per (b, s) group.
// Streams n points in index order; wave-ballot ordered append implements
// "first k indices within radius" without materializing (B,S,N) distances.
// Writes grouped feature rows [feat(F), xyz - center(3), zero-pad] with padded
// row stride CWp so the GEMM kernel needs no bounds checks.
// ============================================================================
__global__ void ballquery_group_kernel(
    const float* __restrict__ cbase, long ccb, long ccp, long ccc,        // coords
    const float* __restrict__ fbase, long fcb, long fcp, long fcc, int F, // point feats
    const float* __restrict__ newXyz, int n, int S, int CWp,
    float ra, float rb, float rc, int ka, int kb, int kc,
    float* __restrict__ outA, float* __restrict__ outB, float* __restrict__ outC)
{
    const int g    = blockIdx.x;      // b * S + s
    const int b    = g / S;
    const int lane = threadIdx.x;     // wave32
    __shared__ int buf[3][8];

    const float* C  = cbase + (long)b * ccb;
    const float* Fp = fbase + (long)b * fcb;
    const float* nx = newXyz + (long)g * 3;
    const float cx = nx[0], cy = nx[1], cz = nx[2];

    const float rr[3] = { ra*ra, rb*rb, rc*rc };
    const int   kk[3] = { ka, kb, kc };
    int cnt[3] = { 0, 0, 0 };
    const unsigned below = (1u << lane) - 1u;

    for (int base0 = 0; base0 < n; base0 += 32) {
        const int j = base0 + lane;
        float d = 1e30f;
        if (j < n) {
            const float dx = C[ccp*j + ccc*0] - cx;
            const float dy = C[ccp*j + ccc*1] - cy;
            const float dz = C[ccp*j + ccc*2] - cz;
            d = dx*dx + dy*dy + dz*dz;
        }
#pragma unroll
        for (int q = 0; q < 3; ++q) {
            const bool pred = (d <= rr[q]);
            const unsigned m = (unsigned)__ballot(pred);
            if (pred) {
                const int pos = cnt[q] + __popc(m & below);
                if (pos < kk[q]) buf[q][pos] = j;
            }
            cnt[q] += __popc(m);
        }
        if (cnt[0] >= kk[0] && cnt[1] >= kk[1] && cnt[2] >= kk[2]) break; // uniform
    }
    __syncthreads();

    float* const outs[3] = { outA, outB, outC };
#pragma unroll
    for (int q = 0; q < 3; ++q) {
        const int k = kk[q];
        for (int e = lane; e < k * CWp; e += 32) {
            const int t = e / CWp, c = e % CWp;
            int idx;
            if (t < cnt[q])      idx = buf[q][t];
            else if (cnt[q] > 0) idx = buf[q][0];   // pad with first (ref semantics)
            else                 idx = n - 1;       // clamp like take_along_axis
            float v = 0.0f;                          // zero-fill K padding
            if (c < F) {
                v = Fp[fcp*idx + fcc*c];
            } else if (c < F + 3) {
                const int a = c - F;
                const float ctr = (a == 0) ? cx : ((a == 1) ? cy : cz);
                v = C[ccp*idx + ccc*a] - ctr;
            }
            outs[q][((long)g * k + t) * CWp + c] = v;
        }
    }
}

// ============================================================================
// Pre-pack a layer's (cin x co) weights into B-fragment lane order, zero-padded
// to (cinP x coP). Layout: block (ct, kt) of 64 floats = [lane][j], value
// B[kt*4 + j + 2*(lane>>4)][ct*16 + (lane&15)]. Also zero-pads s/b so padded
// output columns are exactly 0 after relu.
// ============================================================================
__global__ void pack_weights_kernel(const float* __restrict__ W, const float* __restrict__ s,
                                    const float* __restrict__ b, float* __restrict__ Wp,
                                    float* __restrict__ sp, float* __restrict__ bp,
                                    int cin, int co, int cinP, int coP)
{
    const int e = blockIdx.x * blockDim.x + threadIdx.x;
    const int wtot = cinP * coP;
    if (e < wtot) {
        const int KT   = cinP >> 2;
        const int blk  = e >> 6;            // 64 floats per fragment block
        const int w    = e & 63;
        const int ct   = blk / KT;
        const int kt   = blk % KT;
        const int lane = w >> 1, j = w & 1;
        const int row  = kt * 4 + j + 2 * (lane >> 4);
        const int col  = ct * 16 + (lane & 15);
        Wp[e] = (row < cin && col < co) ? W[row * co + col] : 0.0f;
    } else if (e < wtot + coP) {
        const int c = e - wtot;
        sp[c] = (c < co) ? s[c] : 0.0f;
        bp[c] = (c < co) ? b[c] : 0.0f;
    }
}

// ============================================================================
// Branchless fused MLP layer: Y = relu((X @ W) * s + b), exact fp32 via
// V_WMMA_F32_16X16X4_F32. One wave per 16x16 tile. rows % 16 == 0,
// cinP % 4 == 0, coP % 16 == 0 guaranteed by construction, so the K loop is
// 2 coalesced b64 loads + 1 wmma per step — no exec-mask juggling.
// ============================================================================
__global__ void wmma_mlp_kernel(const float* __restrict__ X, const float* __restrict__ Wp,
                                const float* __restrict__ sp, const float* __restrict__ bp,
                                float* __restrict__ Y, int cinP, int coP)
{
    const int lane = threadIdx.x;     // 0..31, wave32
    const int half = lane >> 4;       // 0/1
    const int l16  = lane & 15;
    const int row0 = blockIdx.x * 16;
    const int ct   = blockIdx.y;
    const int KT   = cinP >> 2;

    const float* xp = X  + (long)(row0 + l16) * cinP + 2 * half;  // A[M=l16][K=.. , ..+1]
    const float* wp = Wp + ((long)ct * KT) * 64 + lane * 2;       // pre-packed B fragments

    v8f acc = {};
    for (int kt = 0; kt < KT; ++kt) {
        const v2f a  = *reinterpret_cast<const v2f*>(xp + 4 * kt);
        const v2f bm = *reinterpret_cast<const v2f*>(wp + 64 * kt);
        acc = __builtin_amdgcn_wmma_f32_16x16x4_f32(
                  false, a, false, bm, (short)0, acc, false, false);
    }
    const int c  = ct * 16 + l16;
    const float sc = sp[c], bb = bp[c];
    float* yp = Y + (long)(row0 + 8 * half) * coP + c;            // C: VGPR j -> M = j + 8*half
#pragma unroll
    for (int j = 0; j < 8; ++j) {
        const float v = acc[j] * sc + bb;
        yp[(long)j * coP] = v > 0.0f ? v : 0.0f;
    }
}

// max over the k grouped samples -> write channel slice of the concat output
__global__ void maxpool_kernel(const float* __restrict__ in, float* __restrict__ out,
                               int G, int k, int co, int inStride, int outStride, int outOff)
{
    const int e = blockIdx.x * blockDim.x + threadIdx.x;
    if (e >= G * co) return;
    const int g = e / co, o = e % co;
    float m = -3.0e38f;
    for (int t = 0; t < k; ++t) m = fmaxf(m, in[(long)(g * k + t) * inStride + o]);
    out[(long)g * outStride + outOff + o] = m;
}

// gp = concat([l2_xyz, l2_pts], -1) padded to 84 cols: rows = B*8
__global__ void concat_sa3_kernel(const float* __restrict__ l2xyz,
                                  const float* __restrict__ l2pts,
                                  float* __restrict__ gp, int rows)
{
    const int e = blockIdx.x * blockDim.x + threadIdx.x;
    if (e >= rows * 84) return;
    const int r = e / 84, c = e % 84;
    float v = 0.0f;
    if (c < 3)       v = l2xyz[r * 3 + c];
    else if (c < 83) v = l2pts[r * 80 + (c - 3)];
    gp[e] = v;
}

// ============================================================================
extern "C" void kernel_launch(void* const* d_in, const int* in_sizes, int n_in,
                              void* d_out, int out_size, void* d_ws, size_t ws_size,
                              hipStream_t stream)
{
    (void)in_sizes; (void)n_in; (void)out_size; (void)ws_size;
    const int B = 16; const long N = 65536;
    const float* xyz = (const float*)d_in[0];

    // Param leaves in setup_inputs() insertion order: per layer {W, s, b};
    // layers 0..8 = sa1 (branch-major), 9..17 = sa2, 18..20 = sa3.
    const float *Wraw[21], *Sraw[21], *Braw[21];
    {
        int p = 1;
        for (int i = 0; i < 21; ++i) {
            Wraw[i] = (const float*)d_in[p + 0];
            Sraw[i] = (const float*)d_in[p + 1];
            Braw[i] = (const float*)d_in[p + 2];
            p += 3;
        }
    }
    // Layer dims (cin, co) and padded dims. cinP of chain head = pad4(cin);
    // inner layers inherit previous coP as their X stride.
    static const int CH[21][2] = {
        {6,4},{4,4},{4,8},  {6,8},{8,8},{8,16},  {6,8},{8,12},{12,16},
        {43,8},{8,8},{8,16},{43,16},{16,16},{16,32},{43,16},{16,16},{16,32},
        {83,32},{32,64},{64,128} };
    int cinP[21], coP[21];
    for (int i = 0; i < 21; ++i) coP[i] = (CH[i][1] + 15) & ~15;
    for (int c0 = 0; c0 < 21; c0 += 3) {
        cinP[c0]     = (CH[c0][0] + 3) & ~3;
        cinP[c0 + 1] = coP[c0];
        cinP[c0 + 2] = coP[c0 + 1];
    }

    // ---- workspace bump allocator (everything stays L2-resident) ----
    char* ws = (char*)d_ws; size_t off = 0;
    auto alloc = [&](size_t bytes) -> void* {
        void* r = ws + off; off += (bytes + 255) & ~(size_t)255; return r;
    };
    float* dist  = (float*)alloc((size_t)B * N * 4);          // FPS scratch (reused)
    int*   cent1 = (int*)  alloc((size_t)B * 32 * 4);
    float* l1xyz = (float*)alloc((size_t)B * 32 * 3 * 4);
    float* g1a   = (float*)alloc((size_t)512  * 8  * 4);      // rows x cinP(=8)
    float* g1b   = (float*)alloc((size_t)1024 * 8  * 4);
    float* g1c   = (float*)alloc((size_t)4096 * 8  * 4);
    float* act0  = (float*)alloc((size_t)4096 * 128 * 4);
    float* act1  = (float*)alloc((size_t)4096 * 128 * 4);
    float* l1pts = (float*)alloc((size_t)B * 32 * 40 * 4);
    int*   cent2 = (int*)  alloc((size_t)B * 8 * 4);
    float* l2xyz = (float*)alloc((size_t)B * 8 * 3 * 4);
    float* h2a   = (float*)alloc((size_t)256  * 44 * 4);      // rows x cinP(=44)
    float* h2b   = (float*)alloc((size_t)512  * 44 * 4);
    float* h2c   = (float*)alloc((size_t)1024 * 44 * 4);
    float* l2pts = (float*)alloc((size_t)B * 8 * 80 * 4);
    float* gp3   = (float*)alloc((size_t)128 * 84 * 4);       // rows x cinP(=84)

    float *Wpk[21], *Spk[21], *Bpk[21];
    for (int i = 0; i < 21; ++i) {
        Wpk[i] = (float*)alloc((size_t)cinP[i] * coP[i] * 4);
        Spk[i] = (float*)alloc((size_t)coP[i] * 4);
        Bpk[i] = (float*)alloc((size_t)coP[i] * 4);
    }
    // ---- pack all weights (tiny, once per launch) ----
    for (int i = 0; i < 21; ++i) {
        const int tot = cinP[i] * coP[i] + coP[i];
        pack_weights_kernel<<<(tot + 255) / 256, 256, 0, stream>>>(
            Wraw[i], Sraw[i], Braw[i], Wpk[i], Spk[i], Bpk[i],
            CH[i][0], CH[i][1], cinP[i], coP[i]);
    }

    auto run_mlp = [&](const float* in, int rows, int base) -> const float* {
        const float* cur = in;
        float* bufs[2] = { act0, act1 };
        for (int l = 0; l < 3; ++l) {
            const int i = base + l;
            float* outp = bufs[l & 1];
            dim3 grid(rows / 16, coP[i] / 16);
            wmma_mlp_kernel<<<grid, 32, 0, stream>>>(cur, Wpk[i], Spk[i], Bpk[i],
                                                     outp, cinP[i], coP[i]);
            cur = outp;
        }
        return cur;
    };
    auto pool = [&](const float* in, float* outp, int G, int k, int co, int inStride,
                    int outStride, int offc) {
        const int tot = G * co;
        maxpool_kernel<<<(tot + 255) / 256, 256, 0, stream>>>(in, outp, G, k, co,
                                                              inStride, outStride, offc);
    };

    // ================= SA1 =================
    fps_kernel<<<B, 1024, 0, stream>>>(xyz, 6 * N, 1, N, (int)N, 32, cent1, l1xyz, dist);
    ballquery_group_kernel<<<B * 32, 32, 0, stream>>>(
        xyz, 6 * N, 1, N,                 // coords: channels 0..2
        xyz + 3 * N, 6 * N, 1, N, 3,      // feats: normals, channels 3..5
        l1xyz, (int)N, 32, 8,
        0.01f, 0.02f, 0.04f, 1, 2, 8, g1a, g1b, g1c);

    { const float* o = run_mlp(g1a,  512, 0); pool(o, l1pts, 512, 1,  8, 16, 40,  0); }
    { const float* o = run_mlp(g1b, 1024, 3); pool(o, l1pts, 512, 2, 16, 16, 40,  8); }
    { const float* o = run_mlp(g1c, 4096, 6); pool(o, l1pts, 512, 8, 16, 16, 40, 24); }

    // ================= SA2 =================
    fps_kernel<<<B, 1024, 0, stream>>>(l1xyz, 96, 3, 1, 32, 8, cent2, l2xyz, dist);
    ballquery_group_kernel<<<B * 8, 32, 0, stream>>>(
        l1xyz, 96, 3, 1,
        l1pts, 1280, 40, 1, 40,
        l2xyz, 32, 8, 44,
        0.02f, 0.04f, 0.08f, 2, 4, 8, h2a, h2b, h2c);

    { const float* o = run_mlp(h2a,  256,  9); pool(o, l2pts, 128, 2, 16, 16, 80,  0); }
    { const float* o = run_mlp(h2b,  512, 12); pool(o, l2pts, 128, 4, 32, 32, 80, 16); }
    { const float* o = run_mlp(h2c, 1024, 15); pool(o, l2pts, 128, 8, 32, 32, 80, 48); }

    // ================= SA3 =================
    concat_sa3_kernel<<<(128 * 84 + 255) / 256, 256, 0, stream>>>(l2xyz, l2pts, gp3, 128);
    { const float* o = run_mlp(gp3, 128, 18);
      pool(o, (float*)d_out, 16, 8, 128, 128, 128, 0); }   // (B, 128) output
}